// DNN_WPE_38139309588796
// MI455X (gfx1250) — compile-verified
//
#include <hip/hip_runtime.h>

typedef float v2f __attribute__((ext_vector_type(2)));
typedef float v8f __attribute__((ext_vector_type(8)));

#define B_      8
#define T_      512
#define C_      8
#define F_      257
#define TAPS    5
#define DELAY   3
#define KD      40      // TAPS * C_
#define NP      48      // padded to 3x16; cols 40..47 carry b (RHS)
#define PAD     8       // leading zero pad: implements Ytil zero-padding branch-free
#define PT      521     // PAD + T_ + 1 (odd-ish stride to de-phase LDS banks)
#define EPSF    1e-10f
#define NWAVES  9
#define NTHREADS 288    // 9 wave32

__global__ __launch_bounds__(NTHREADS)
void wpe_kernel(const float* __restrict__ dre,
                const float* __restrict__ dim_,
                float* __restrict__ out)
{
    // ~105 KB of the 320 KB WGP pool
    __shared__ float2 Yc[C_][PT];                 // interleaved (re,im), data at [c][PAD+t]
    __shared__ float2 Ec[C_][PT];                 // enhanced, same layout
    __shared__ float  wts[T_];                    // per-frame weights (0 for t<7)
    __shared__ float  Mre[NP][NP], Mim[NP][NP];   // augmented [R+epsI | b]; rows 40..47 = 0
    __shared__ float  tileP[NWAVES][2][16][16];   // per-wave WMMA tile scratch

    const int bf   = blockIdx.x;        // 0 .. B_*F_-1
    const int b    = bf / F_;
    const int f    = bf % F_;
    const int tid  = threadIdx.x;
    const int lane = tid & 31;
    const int wave = tid >> 5;
    const int rlo  = lane & 15;         // M (A) / N (B,C,D) index within tile
    const int khi  = lane >> 4;         // K half-select for 16x16x4 layout

    // ---- zero the leading pad and the pad rows of M (stay zero forever) ----
    for (int i = tid; i < C_ * PAD; i += NTHREADS) {
        Yc[i >> 3][i & 7] = make_float2(0.f, 0.f);
        Ec[i >> 3][i & 7] = make_float2(0.f, 0.f);
    }
    for (int i = tid; i < (NP - KD) * NP; i += NTHREADS) {
        Mre[KD + i / NP][i % NP] = 0.f;
        Mim[KD + i / NP][i % NP] = 0.f;
    }
    // ---- load Y(b, :, :, f); adjacent blocks share cachelines in f ----
    for (int i = tid; i < C_ * T_; i += NTHREADS) {
        int c = i >> 9;             // / T_
        int t = i & (T_ - 1);
        size_t g = ((size_t)(b * T_ + t) * C_ + c) * F_ + f;
        Yc[c][PAD + t] = make_float2(dre[g], dim_[g]);
    }
    __syncthreads();

    // stage-D A-side pointer: rows 0..7 = G_re^T, rows 8..15 = G_im^T (lane-fixed)
    const float* gptr = (rlo < 8) ? &Mre[0][KD + rlo] : &Mim[0][KD + rlo - 8];

    for (int iter = 0; iter < 2; ++iter) {
        // ================= power -> weights =================
        {
            const float2 (*Sc)[PT] = (iter == 0) ? Yc : Ec;
            for (int t = tid; t < T_; t += NTHREADS) {
                float p = 0.f;
                #pragma unroll
                for (int c = 0; c < C_; ++c) {
                    float2 v = Sc[c][PAD + t];
                    p += v.x * v.x + v.y * v.y;
                }
                p *= (1.0f / C_);
                wts[t] = (t >= DELAY + TAPS - 1) ? 1.0f / fmaxf(p, EPSF) : 0.0f;
            }
        }
        __syncthreads();

        // ======== Stage B: [R | b] via f32 WMMA, branch-free K-loop ========
        {
            const int mt = wave / 3, nt = wave % 3;
            const int am = mt * 16 + rlo;               // A row (0..47; >=KD are pad rows)
            const int bn = nt * 16 + rlo;               // B col (0..47)

            // per-lane base pointers (leading pad keeps every access in-bounds)
            const float2* aptr = &Yc[am & 7][PAD - DELAY - (am >> 3)];
            const int bshift   = (bn < KD) ? (DELAY + (bn >> 3)) : 0;
            const int bchan    = (bn < KD) ? (bn & 7) : (bn - KD);
            const float2* bptr = &Yc[bchan][PAD - bshift];
            const float2* wp   = (const float2*)wts;

            v8f arr = {}, ari = {}, air = {}, aii = {};

            #pragma unroll 4
            for (int k0 = 0; k0 < T_; k0 += 4) {
                const int t0 = k0 + 2 * khi;
                float2 w  = wp[(k0 >> 1) + khi];        // w==0 masks all t<7 terms
                float2 a0 = aptr[t0], a1 = aptr[t0 + 1];
                float2 b0 = bptr[t0], b1 = bptr[t0 + 1];
                v2f Are = { w.x * a0.x, w.y * a1.x };
                v2f Aim = { w.x * a0.y, w.y * a1.y };
                v2f Bre = { b0.x, b1.x };
                v2f Bim = { b0.y, b1.y };
                arr = __builtin_amdgcn_wmma_f32_16x16x4_f32(false, Are, false, Bre, (short)0, arr, false, false);
                ari = __builtin_amdgcn_wmma_f32_16x16x4_f32(false, Are, false, Bim, (short)0, ari, false, false);
                air = __builtin_amdgcn_wmma_f32_16x16x4_f32(false, Aim, false, Bre, (short)0, air, false, false);
                aii = __builtin_amdgcn_wmma_f32_16x16x4_f32(false, Aim, false, Bim, (short)0, aii, false, false);
            }

            // conj(A)*B combine + epsI, straight into augmented matrix
            #pragma unroll
            for (int r = 0; r < 8; ++r) {
                int row = mt * 16 + r + 8 * khi;
                int col = nt * 16 + rlo;
                if (row < KD) {
                    float rre = arr[r] + aii[r];
                    float rim = ari[r] - air[r];
                    if (row == col) rre += EPSF;
                    Mre[row][col] = rre;
                    Mim[row][col] = rim;
                }
            }
        }
        __syncthreads();

        // ======== Stage C: complex Gaussian elimination (HPD, no pivot) ========
        for (int k = 0; k < KD; ++k) {
            float pr = Mre[k][k], pi = Mim[k][k];
            float inv = 1.0f / (pr * pr + pi * pi);
            float qr = pr * inv, qi = -pi * inv;        // 1/pivot
            for (int j = k + 1 + tid; j < NP; j += NTHREADS) {
                float ar = Mre[k][j], ai = Mim[k][j];
                Mre[k][j] = ar * qr - ai * qi;
                Mim[k][j] = ar * qi + ai * qr;
            }
            __syncthreads();
            const int ncol = NP - k - 1;
            for (int idx = tid; idx < (KD - 1 - k) * ncol; idx += NTHREADS) {
                int i = k + 1 + idx / ncol;
                int j = k + 1 + idx % ncol;
                float fr = Mre[i][k], fi = Mim[i][k];
                float ar = Mre[k][j], ai = Mim[k][j];
                Mre[i][j] -= fr * ar - fi * ai;
                Mim[i][j] -= fr * ai + fi * ar;
            }
            __syncthreads();
        }
        // back substitution on the 8 RHS columns -> G lands in M[:, 40..47]
        for (int k = KD - 1; k >= 1; --k) {
            for (int idx = tid; idx < k * C_; idx += NTHREADS) {
                int i = idx >> 3;
                int j = KD + (idx & 7);
                float fr = Mre[i][k], fi = Mim[i][k];
                float ar = Mre[k][j], ai = Mim[k][j];
                Mre[i][j] -= fr * ar - fi * ai;
                Mim[i][j] -= fr * ai + fi * ar;
            }
            __syncthreads();
        }

        // ======== Stage D: reverb tail GEMM + enhanced = Y - tail ========
        for (int ntile = wave; ntile < T_ / 16; ntile += NWAVES) {
            const int t0 = ntile * 16;
            const int tn = t0 + rlo;            // time column for B fragments
            v8f c1 = {}, c2 = {};
            #pragma unroll
            for (int k0 = 0; k0 < NP; k0 += 4) {
                v2f Ag, Br, Bi;
                #pragma unroll
                for (int u = 0; u < 2; ++u) {
                    const int pd = k0 + 2 * khi + u;
                    float2 y = Yc[pd & 7][PAD + tn - DELAY - (pd >> 3)];
                    Ag[u] = gptr[pd * NP];      // pad rows (pd>=KD) are zero
                    Br[u] = y.x;
                    Bi[u] = y.y;
                }
                c1 = __builtin_amdgcn_wmma_f32_16x16x4_f32(false, Ag, false, Br, (short)0, c1, false, false);
                c2 = __builtin_amdgcn_wmma_f32_16x16x4_f32(false, Ag, false, Bi, (short)0, c2, false, false);
            }
            // stage through wave-private LDS for cross-lane complex recombine
            #pragma unroll
            for (int r = 0; r < 8; ++r) {
                tileP[wave][0][r + 8 * khi][rlo] = c1[r];
                tileP[wave][1][r + 8 * khi][rlo] = c2[r];
            }
            #pragma unroll
            for (int idx = lane; idx < 128; idx += 32) {
                int e = idx >> 4;
                int n = idx & 15;
                int t = t0 + n;
                float p1e = tileP[wave][0][e][n];       // G_re . Yt_re
                float p1i = tileP[wave][0][8 + e][n];   // G_im . Yt_re
                float p2e = tileP[wave][1][e][n];       // G_re . Yt_im
                float p2i = tileP[wave][1][8 + e][n];   // G_im . Yt_im
                float2 y = Yc[e][PAD + t];
                Ec[e][PAD + t] = make_float2(y.x - (p1e - p2i),
                                             y.y - (p2e + p1i));
            }
        }
        __syncthreads();
    } // iterations

    // ---- write enhanced (B,T,C,F,2) ----
    for (int i = tid; i < C_ * T_; i += NTHREADS) {
        int c = i >> 9;
        int t = i & (T_ - 1);
        size_t g = ((size_t)(b * T_ + t) * C_ + c) * F_ + f;
        float2 e = Ec[c][PAD + t];
        out[2 * g + 0] = e.x;
        out[2 * g + 1] = e.y;
    }
}

extern "C" void kernel_launch(void* const* d_in, const int* in_sizes, int n_in,
                              void* d_out, int out_size, void* d_ws, size_t ws_size,
                              hipStream_t stream) {
    (void)in_sizes; (void)n_in; (void)d_ws; (void)ws_size; (void)out_size;
    const float* dre  = (const float*)d_in[0];
    const float* dimg = (const float*)d_in[1];
    // d_in[2] = ilens, unused (all == T)
    float* out = (float*)d_out;
    dim3 grid(B_ * F_);
    dim3 block(NTHREADS);
    hipLaunchKernelGGL(wpe_kernel, grid, block, 0, stream, dre, dimg, out);
}